// VoxelPooler_7121055777582
// MI455X (gfx1250) — compile-verified
//
#include <hip/hip_runtime.h>
#include <hip/hip_bf16.h>
#include <stdint.h>

// Voxel scatter-add pooling for MI455X (gfx1250).
//
// Memory-bound scatter workload: no matmul structure -> WMMA not applicable.
// CDNA5-specific path used instead: double-buffered GLOBAL_LOAD_ASYNC_TO_LDS_B128
// (ASYNCcnt) staging of each point's 256B feature vector, overlapped with
// hardware global_atomic_add_f32 scatter (no-return form, STOREcnt).

#define GX 200
#define GY 200
#define GZ 8
#define CCH 64
#define XY (GX * GY)
#define WAVES_PER_BLOCK 8

// Issue one async 16B copy per lane (lanes 0..15) from global to LDS.
// One instruction per wave -> ASYNCcnt increments by exactly 1 per stage.
__device__ __forceinline__ void stage_async(const float* __restrict__ src,
                                            unsigned lds_byte_off,
                                            bool active) {
    if (active) {
        asm volatile("global_load_async_to_lds_b128 %0, %1, off"
                     :
                     : "v"(lds_byte_off), "v"(src)
                     : "memory");
    }
}

__device__ __forceinline__ void atomic_add_f32_noret(float* addr, float val) {
    asm volatile("global_atomic_add_f32 %0, %1, off"
                 :
                 : "v"(addr), "v"(val)
                 : "memory");
}

__global__ void __launch_bounds__(256)
voxel_scatter_kernel(const float* __restrict__ geom,   // [totalPts, 3]
                     const float* __restrict__ feat,   // [totalPts, 64]
                     const float* __restrict__ vsize,  // [3]
                     const float* __restrict__ vorig,  // [3]
                     float* __restrict__ out,          // [B, GZ*64, GX, GY]
                     int totalPts, int ptsPerBatch) {
    // Double-buffered per-wave staging: 2 slots x 8 waves x 64 floats = 4 KB.
    __shared__ __align__(16) float lds[2][WAVES_PER_BLOCK][CCH];

    const int lane = threadIdx.x & 31;
    const int wv   = threadIdx.x >> 5;
    const int wgl  = blockIdx.x * WAVES_PER_BLOCK + wv;
    const int stride = gridDim.x * WAVES_PER_BLOCK;

    const float ox = vorig[0], oy = vorig[1], oz = vorig[2];
    const float sx = vsize[0], sy = vsize[1], sz = vsize[2];

    int p = wgl;
    int slot = 0;

    // Prologue: stage first point's features into slot 0.
    if (p < totalPts) {
        stage_async(feat + (size_t)p * CCH + lane * 4,
                    (unsigned)(uintptr_t)&lds[0][wv][lane * 4],
                    lane < 16);
    }

    for (; p < totalPts; p += stride) {
        const int pn = p + stride;
        if (pn < totalPts) {
            // Stage next point into the other slot, then wait for current slot.
            stage_async(feat + (size_t)pn * CCH + lane * 4,
                        (unsigned)(uintptr_t)&lds[slot ^ 1][wv][lane * 4],
                        lane < 16);
            asm volatile("s_wait_asynccnt 0x1" ::: "memory");
        } else {
            asm volatile("s_wait_asynccnt 0x0" ::: "memory");
        }

        // Wave-uniform voxel index computation (all lanes redundantly compute).
        const size_t g3 = (size_t)p * 3;
        const float gx = geom[g3 + 0];
        const float gy = geom[g3 + 1];
        const float gz = geom[g3 + 2];
        const int ix = (int)floorf((gx - ox) / sx);
        const int iy = (int)floorf((gy - oy) / sy);
        const int iz = (int)floorf((gz - oz) / sz);

        const bool valid = (ix >= 0) & (ix < GX) &
                           (iy >= 0) & (iy < GY) &
                           (iz >= 0) & (iz < GZ);
        if (valid) {
            const int b = p / ptsPerBatch;
            // out[b, iz*64 + c, ix, iy], channels are XY-plane strided.
            const size_t base = ((size_t)b * (GZ * CCH) + (size_t)iz * CCH) * XY
                              + (size_t)ix * GY + (size_t)iy;
            const int c = lane * 2;
            const float2 v = *(const float2*)&lds[slot][wv][c];
            atomic_add_f32_noret(out + base + (size_t)c * XY,       v.x);
            atomic_add_f32_noret(out + base + (size_t)(c + 1) * XY, v.y);
        }
        slot ^= 1;
    }
}

extern "C" void kernel_launch(void* const* d_in, const int* in_sizes, int n_in,
                              void* d_out, int out_size, void* d_ws, size_t ws_size,
                              hipStream_t stream) {
    const float* geom  = (const float*)d_in[0];
    const float* feat  = (const float*)d_in[1];
    const float* vsize = (const float*)d_in[2];
    const float* vorig = (const float*)d_in[3];
    float* out = (float*)d_out;

    const int totalPts = in_sizes[0] / 3;                    // B*N*D*H*W
    const int B = out_size / (GZ * CCH * XY);                // batches
    const int ptsPerBatch = (B > 0) ? (totalPts / B) : totalPts;

    // Zero the dense voxel grid (graph-capturable memset node).
    hipMemsetAsync(d_out, 0, (size_t)out_size * sizeof(float), stream);

    // Persistent-style grid: 16384 waves, each pipelines ~42 points.
    const int blocks = 2048;
    voxel_scatter_kernel<<<blocks, 256, 0, stream>>>(
        geom, feat, vsize, vorig, out, totalPts, ptsPerBatch);
}